// YOLOLayer_10196252360956
// MI455X (gfx1250) — compile-verified
//
#include <hip/hip_runtime.h>
#include <math.h>

// ---- problem constants (fixed by the reference harness) ----
#define NB 16
#define NA 3
#define NH 128
#define NW 128
#define NC 80
#define STRIDE_PX 8.0f
#define NCELLS (NB * NA * NH * NW)     // 786432
#define TILE 128                       // cells per block (1 cell / thread)
#define LDS_STRIDE 84                  // dwords per cell: 80 + 4 pad (336B, 16B aligned)
#define CHUNKS_PER_CELL 20             // 20 x float4 = 80 floats

#if defined(__has_builtin)
#if __has_builtin(__builtin_amdgcn_global_load_async_to_lds_b128) && \
    __has_builtin(__builtin_amdgcn_s_wait_asynccnt)
#define HAS_ASYNC_LDS 1
#endif
#endif
#ifndef HAS_ASYNC_LDS
#define HAS_ASYNC_LDS 0
#endif

typedef int v4i __attribute__((ext_vector_type(4)));
typedef __attribute__((address_space(1))) v4i* gv4i_p;  // global int4*
typedef __attribute__((address_space(3))) v4i* lv4i_p;  // LDS int4*

__device__ __forceinline__ float fsigmoid(float x) {
    return 1.0f / (1.0f + __expf(-x));
}

__global__ __launch_bounds__(TILE) void yolo_decode_kernel(
    const float* __restrict__ t_xywha,    // [NCELLS, 5]
    const float* __restrict__ conf_logits,// [NCELLS]
    const float* __restrict__ cls_logits, // [NCELLS, 80]
    const float* __restrict__ anchors,    // [NA, 2]
    float* __restrict__ out,              // [NCELLS*5 | NCELLS | NCELLS]
    int ncells)
{
    __shared__ float sh[TILE * LDS_STRIDE];  // 43,008 B

    const int  t     = threadIdx.x;
    const int  tile0 = (int)blockIdx.x * TILE;       // first cell of this tile
    if (tile0 >= ncells) return;
    const long cls_base = (long)tile0 * NC;          // float index into cls_logits

    // ---------------- stage class-logit tile into LDS ----------------
    // Chunk j (16B) of the tile: cell c = j/20, channel k4 = (j%20)*4.
    // Consecutive lanes take consecutive j -> each instruction streams a
    // contiguous 512B from HBM/L2 into LDS via the async path (no VGPR bounce).
#if HAS_ASYNC_LDS
    #pragma unroll
    for (int it = 0; it < CHUNKS_PER_CELL; ++it) {
        const int j  = t + it * TILE;                // [0, 2560)
        const int c  = j / CHUNKS_PER_CELL;
        const int k4 = (j - c * CHUNKS_PER_CELL) * 4;
        const void* g = (const void*)(cls_logits + cls_base + (long)j * 4);
        void*       l = (void*)&sh[c * LDS_STRIDE + k4];
        __builtin_amdgcn_global_load_async_to_lds_b128(
            (gv4i_p)g, (lv4i_p)l, /*offset=*/0, /*cpol=*/0);
    }
    __builtin_amdgcn_s_wait_asynccnt(0);
#else
    #pragma unroll
    for (int it = 0; it < CHUNKS_PER_CELL; ++it) {
        const int j  = t + it * TILE;
        const int c  = j / CHUNKS_PER_CELL;
        const int k4 = (j - c * CHUNKS_PER_CELL) * 4;
        const float4 v = *(const float4*)(cls_logits + cls_base + (long)j * 4);
        *(float4*)&sh[c * LDS_STRIDE + k4] = v;
    }
#endif
    __syncthreads();

    const int i = tile0 + t;                         // global cell index
    if (i >= ncells) return;

    // ---------------- max / argmax over 80 class logits ----------------
    // sigmoid is strictly monotone: argmax(sigmoid(x)) == argmax(x) and
    // max(sigmoid(x)) == sigmoid(max(x)); first-max tie-break preserved.
    const float* base = &sh[t * LDS_STRIDE];
    float best = -INFINITY;
    int   bidx = 0;
    #pragma unroll
    for (int k = 0; k < NC; k += 4) {
        const float4 v = *(const float4*)(base + k); // ds_load_b128
        if (v.x > best) { best = v.x; bidx = k + 0; }
        if (v.y > best) { best = v.y; bidx = k + 1; }
        if (v.z > best) { best = v.z; bidx = k + 2; }
        if (v.w > best) { best = v.w; bidx = k + 3; }
    }

    // ---------------- decode box ----------------
    const int rem = i % (NA * NH * NW);
    const int a   = rem / (NH * NW);
    const int hw  = rem % (NH * NW);
    const int h   = hw / NW;
    const int w   = hw - h * NW;

    const long i5 = (long)i * 5;
    const float tx = t_xywha[i5 + 0];
    const float ty = t_xywha[i5 + 1];
    const float tw = t_xywha[i5 + 2];
    const float th = t_xywha[i5 + 3];
    const float ta = t_xywha[i5 + 4];

    const float ax = anchors[a * 2 + 0];
    const float ay = anchors[a * 2 + 1];

    const float px   = (fsigmoid(tx) + (float)w) * STRIDE_PX;  // mesh_x = w index
    const float py   = (fsigmoid(ty) + (float)h) * STRIDE_PX;  // mesh_y = h index
    const float pw   = __expf(tw) * ax;
    const float ph   = __expf(th) * ay;
    // (sigmoid(ta)*2pi - pi) / pi * 180 == sigmoid(ta)*360 - 180
    const float pdeg = fsigmoid(ta) * 360.0f - 180.0f;

    const float pconf = fsigmoid(conf_logits[i]);
    const float confv = pconf * fsigmoid(best);

    // ---------------- write concatenated outputs ----------------
    float* oxy = out + i5;
    oxy[0] = px; oxy[1] = py; oxy[2] = pw; oxy[3] = ph; oxy[4] = pdeg;

    out[(long)ncells * 5 + i] = (float)bidx;   // cls_idx (as f32 value)
    out[(long)ncells * 6 + i] = confv;         // confs
}

extern "C" void kernel_launch(void* const* d_in, const int* in_sizes, int n_in,
                              void* d_out, int out_size, void* d_ws, size_t ws_size,
                              hipStream_t stream) {
    const float* t_xywha     = (const float*)d_in[0];
    const float* conf_logits = (const float*)d_in[1];
    const float* cls_logits  = (const float*)d_in[2];
    const float* anchors     = (const float*)d_in[3];
    float*       out         = (float*)d_out;

    const int ncells = in_sizes[1];                 // conf_logits flat count = nB*nA*nH*nW
    const int blocks = (ncells + TILE - 1) / TILE;  // 6144 for the reference shapes

    yolo_decode_kernel<<<dim3(blocks), dim3(TILE), 0, stream>>>(
        t_xywha, conf_logits, cls_logits, anchors, out, ncells);
}